// SSGC_learnable_emb_Module_67362267070695
// MI455X (gfx1250) — compile-verified
//
#include <hip/hip_runtime.h>

typedef __attribute__((ext_vector_type(2))) float v2f;
typedef __attribute__((ext_vector_type(8))) float v8f;

#define N0 2560000
#define N1 160000
#define N2 10000
#define E0 2560000
#define E1 160000
#define D  64

// ---------------- zero workspace ----------------
__global__ void zero_ws_kernel(float4* __restrict__ ws, int n4) {
    int i = blockIdx.x * blockDim.x + threadIdx.x;
    if (i < n4) ws[i] = make_float4(0.f, 0.f, 0.f, 0.f);
}

// ---------------- SPMM: out[dst] += x[src] * w ----------------
// 16 threads per edge, each handles a float4 chunk of the 64-wide row.
// Scatter via native global_atomic_add_f32 (no return) -> lands in L2.
__global__ void spmm_kernel(const float* __restrict__ x,
                            const int* __restrict__ src,
                            const int* __restrict__ dst,
                            const float* __restrict__ ew,
                            float* __restrict__ out,
                            int nEdge) {
    long long idx = (long long)blockIdx.x * blockDim.x + threadIdx.x;
    if (idx >= (long long)nEdge * 16) return;
    int e = (int)(idx >> 4);
    int c = ((int)idx & 15) * 4;
    int s = src[e];
    int d = dst[e];
    float w = ew[e];
    const float4 v = *(const float4*)(x + (long long)s * D + c);
    float* o = out + (long long)d * D + c;
    unsafeAtomicAdd(o + 0, v.x * w);
    unsafeAtomicAdd(o + 1, v.y * w);
    unsafeAtomicAdd(o + 2, v.z * w);
    unsafeAtomicAdd(o + 3, v.w * w);
}

// ---------------- final: h = (x + x1 + x2)/3 ; out = h @ W^T + b ----------------
// One wave per 16-row M tile, 4 N-tiles of 16, K=64 via 16 steps of
// V_WMMA_F32_16X16X4_F32. h staged in LDS per wave, W staged per block.
__global__ void final_gemm_kernel(const float* __restrict__ x,
                                  const float* __restrict__ x1,
                                  const float* __restrict__ x2,
                                  const float* __restrict__ W,
                                  const float* __restrict__ bias,
                                  float* __restrict__ out) {
    __shared__ float Wlds[64 * 64];        // 16 KB
    __shared__ float hlds[8][16 * 64];     // 32 KB (one 16x64 tile per wave)

    const int tid  = threadIdx.x;
    const int wave = tid >> 5;
    const int lane = tid & 31;

    // cooperative W load (64*64 floats = 1024 float4, 256 threads -> 4 each)
    for (int i = tid; i < 64 * 64 / 4; i += 256) {
        ((float4*)Wlds)[i] = ((const float4*)W)[i];
    }

    const int mtile  = blockIdx.x * 8 + wave;
    const bool active = (mtile < (N2 / 16));   // wave-uniform

    if (active) {
        // stage h tile: 16 rows x 64 cols = 256 float4, 32 lanes x 8 each
        const long long base = (long long)mtile * 16 * D;
        const float4* xb  = (const float4*)(x  + base);
        const float4* x1b = (const float4*)(x1 + base);
        const float4* x2b = (const float4*)(x2 + base);
        float4* hl = (float4*)hlds[wave];
        const float inv3 = 1.0f / 3.0f;
        for (int i = lane; i < 256; i += 32) {
            float4 a = xb[i];
            float4 b = x1b[i];
            float4 c = x2b[i];
            float4 h;
            h.x = (a.x + b.x + c.x) * inv3;
            h.y = (a.y + b.y + c.y) * inv3;
            h.z = (a.z + b.z + c.z) * inv3;
            h.w = (a.w + b.w + c.w) * inv3;
            hl[i] = h;
        }
    }
    __syncthreads();
    if (!active) return;

    const int hi = lane >> 4;   // half-wave selector
    const int lo = lane & 15;   // row (A) / col (B,D) index
    const float* hrow = hlds[wave];

    for (int nt = 0; nt < 4; ++nt) {
        const int nBase = nt * 16;
        v8f acc = {};
        for (int kk = 0; kk < 16; ++kk) {
            const int kb = kk * 4 + 2 * hi;
            // A (16x4 f32): lane lo = row m, VGPR0/1 = K kb / kb+1
            v2f a = *(const v2f*)(hrow + lo * 64 + kb);
            // B (4x16 f32) = W^T tile: B[k][n] = W[n][k]
            v2f b = *(const v2f*)(Wlds + (nBase + lo) * 64 + kb);
            acc = __builtin_amdgcn_wmma_f32_16x16x4_f32(
                false, a, false, b, (short)0, acc, false, false);
        }
        const float bv = bias[nBase + lo];
        for (int r = 0; r < 8; ++r) {
            const int m = mtile * 16 + r + 8 * hi;
            out[m * 64 + nBase + lo] = acc[r] + bv;
        }
    }
}

extern "C" void kernel_launch(void* const* d_in, const int* in_sizes, int n_in,
                              void* d_out, int out_size, void* d_ws, size_t ws_size,
                              hipStream_t stream) {
    const float* x      = (const float*)d_in[0];
    const float* W      = (const float*)d_in[1];
    const float* b      = (const float*)d_in[2];
    const float* e0_w   = (const float*)d_in[3];
    const float* e1_w   = (const float*)d_in[4];
    const int*   e0_src = (const int*)d_in[5];
    const int*   e0_dst = (const int*)d_in[6];
    const int*   e1_src = (const int*)d_in[7];
    const int*   e1_dst = (const int*)d_in[8];
    float* out = (float*)d_out;

    float* x1 = (float*)d_ws;                 // [N1, 64] = 40.96 MB
    float* x2 = x1 + (size_t)N1 * D;          // [N2, 64] =  2.56 MB

    // zero x1 and x2 (contiguous in ws)
    const int n4 = (N1 + N2) * D / 4;
    zero_ws_kernel<<<(n4 + 255) / 256, 256, 0, stream>>>((float4*)d_ws, n4);

    // block 0: x1 = scatter_sum(x[e0_src] * e0_w)
    {
        long long nThreads = (long long)E0 * 16;
        int blocks = (int)((nThreads + 255) / 256);
        spmm_kernel<<<blocks, 256, 0, stream>>>(x, e0_src, e0_dst, e0_w, x1, E0);
    }
    // block 1: x2 = scatter_sum(x1[e1_src] * e1_w)
    {
        long long nThreads = (long long)E1 * 16;
        int blocks = (int)((nThreads + 255) / 256);
        spmm_kernel<<<blocks, 256, 0, stream>>>(x1, e1_src, e1_dst, e1_w, x2, E1);
    }
    // final: h = mean(x[:N2], x1[:N2], x2) ; out = h @ W^T + b   (WMMA f32)
    {
        int mtiles = N2 / 16;                 // 625
        int blocks = (mtiles + 7) / 8;        // 8 waves per block
        final_gemm_kernel<<<blocks, 256, 0, stream>>>(x, x1, x2, W, b, out);
    }
}